// SlotAttentionModule_36180804501654
// MI455X (gfx1250) — compile-verified
//
#include <hip/hip_runtime.h>
#include <math.h>
#include <stdint.h>

typedef __attribute__((ext_vector_type(2))) float v2f;
typedef __attribute__((ext_vector_type(8))) float v8f;

#define DD 64
#define KS 8
#define NN 4096
#define BBATCH 128
#define LN_EPS 1e-5f
#define EPS_ATTN 1e-8f
#define INV_SCALE 0.125f   // 1/sqrt(64)

// ---- CDNA5 f32 WMMA: D(16x16) = A(16x4) * B(4x16) + C ----
static __device__ __forceinline__ v8f wmma_f32(v2f a, v2f b, v8f c) {
  return __builtin_amdgcn_wmma_f32_16x16x4_f32(
      /*neg_a=*/false, a, /*neg_b=*/false, b,
      /*c_mod=*/(short)0, c, /*reuse_a=*/false, /*reuse_b=*/false);
}

// ---- CDNA5 async Global->LDS copy (B128, GVS mode), tracked by ASYNCcnt ----
// Low 32 bits of a generic __shared__ pointer are the wave-relative LDS byte
// address (ISA: LDS aperture check truncates to addr[31:0]).
static __device__ __forceinline__ uint32_t lds_addr32(const void* p) {
  return (uint32_t)(uintptr_t)p;
}
static __device__ __forceinline__ void async_load_b128(uint32_t lds_dst,
                                                       uint32_t goff,
                                                       uint64_t base) {
  asm volatile("global_load_async_to_lds_b128 %0, %1, %2"
               :: "v"(lds_dst), "v"(goff), "s"(base)
               : "memory");
}
static __device__ __forceinline__ void async_wait0() {
  asm volatile("s_wait_asynccnt 0" ::: "memory");
}

// =====================================================================
// Kernel 0: slots0 = slot_mu + exp(slot_log_sigma) * noise
// =====================================================================
__global__ void init_slots_kernel(const float* __restrict__ noise,
                                  const float* __restrict__ mu,
                                  const float* __restrict__ ls,
                                  float* __restrict__ s0, int n) {
  int i = blockIdx.x * 256 + threadIdx.x;
  if (i < n) {
    int d = i & (DD - 1);
    s0[i] = mu[d] + expf(ls[d]) * noise[i];
  }
}

__global__ void zero_kernel(float* __restrict__ p, int n) {
  int i = blockIdx.x * 256 + threadIdx.x;
  if (i < n) p[i] = 0.0f;
}

// =====================================================================
// Kernel 1: x = LN(inputs); k = x @ Wk^T; v = x @ Wv^T   (WMMA f32)
// 128 rows per block (8 waves x 16-row WMMA tiles). All LDS staging via
// async Global->LDS B128 copies. Two passes over the shared weight
// buffer (Wk then Wv) to stay under 64KB static LDS.
// =====================================================================
#define LNKV_ROWS 128
#define XS_STR 68   // stride-68: conflict-free float2 column reads, 16B-aligned rows

__global__ __launch_bounds__(256) void ln_kv_kernel(
    const float* __restrict__ inp, const float* __restrict__ lng,
    const float* __restrict__ lnb, const float* __restrict__ Wk,
    const float* __restrict__ Wv, float* __restrict__ kout,
    float* __restrict__ vout) {
  __shared__ float xs[LNKV_ROWS * XS_STR];   // 34816 B
  __shared__ float ws[DD * XS_STR];          // 17408 B (Wk, then Wv)
  const int t = threadIdx.x;
  const int lane = t & 31, wave = t >> 5;
  const int m16 = lane & 15, hi = lane >> 4, h2 = hi << 1;
  const long rowBase = (long)blockIdx.x * LNKV_ROWS;

  // async-stage 128x64 input tile (2048 float4) + Wk (1024 float4)
  {
    const uint64_t xbase = (uint64_t)(inp + rowBase * DD);
    for (int i = 0; i < 8; ++i) {
      int fidx = i * 256 + t, r = fidx >> 4, c4 = fidx & 15;
      async_load_b128(lds_addr32(&xs[r * XS_STR + c4 * 4]),
                      (uint32_t)(fidx * 16), xbase);
    }
    const uint64_t wbase = (uint64_t)Wk;
    for (int i = 0; i < 4; ++i) {
      int fidx = i * 256 + t, r = fidx >> 4, c4 = fidx & 15;
      async_load_b128(lds_addr32(&ws[r * XS_STR + c4 * 4]),
                      (uint32_t)(fidx * 16), wbase);
    }
    async_wait0();       // per-wave ASYNCcnt -> 0 before making data visible
  }
  __syncthreads();

  // per-row LayerNorm (one thread per row)
  if (t < LNKV_ROWS) {
    float* row = &xs[t * XS_STR];
    float m = 0.0f;
    for (int c = 0; c < DD; ++c) m += row[c];
    m *= (1.0f / DD);
    float va = 0.0f;
    for (int c = 0; c < DD; ++c) { float d = row[c] - m; va += d * d; }
    va *= (1.0f / DD);
    float rstd = rsqrtf(va + LN_EPS);
    for (int c = 0; c < DD; ++c) row[c] = (row[c] - m) * rstd * lng[c] + lnb[c];
  }
  __syncthreads();

  // A fragments: lane holds row m16, cols {4s+h2, 4s+h2+1}  (16x4 f32 layout)
  const int wr = wave * 16;
  v2f a[16];
  for (int s = 0; s < 16; ++s)
    a[s] = *(const v2f*)&xs[(wr + m16) * XS_STR + 4 * s + h2];

  const long orow0 = rowBase + wr + hi * 8;
  // pass 1: k = x @ Wk^T
  for (int nt = 0; nt < 4; ++nt) {
    v8f c = {};
    for (int s = 0; s < 16; ++s) {
      v2f b = *(const v2f*)&ws[(nt * 16 + m16) * XS_STR + 4 * s + h2];
      c = wmma_f32(a[s], b, c);
    }
    for (int j = 0; j < 8; ++j)
      kout[(orow0 + j) * DD + nt * 16 + m16] = c[j];
  }
  __syncthreads();
  {
    const uint64_t wbase = (uint64_t)Wv;
    for (int i = 0; i < 4; ++i) {
      int fidx = i * 256 + t, r = fidx >> 4, c4 = fidx & 15;
      async_load_b128(lds_addr32(&ws[r * XS_STR + c4 * 4]),
                      (uint32_t)(fidx * 16), wbase);
    }
    async_wait0();
  }
  __syncthreads();
  // pass 2: v = x @ Wv^T
  for (int nt = 0; nt < 4; ++nt) {
    v8f c = {};
    for (int s = 0; s < 16; ++s) {
      v2f b = *(const v2f*)&ws[(nt * 16 + m16) * XS_STR + 4 * s + h2];
      c = wmma_f32(a[s], b, c);
    }
    for (int j = 0; j < 8; ++j)
      vout[(orow0 + j) * DD + nt * 16 + m16] = c[j];
  }
}

// =====================================================================
// Kernel 2: q = LN(slots) @ Wq^T, padded to 16 slot rows (rows 8..15 = 0)
// =====================================================================
__global__ __launch_bounds__(256) void q_kernel(
    const float* __restrict__ slots, const float* __restrict__ lng,
    const float* __restrict__ lnb, const float* __restrict__ Wq,
    float* __restrict__ qpad) {
  __shared__ float sn[KS * DD];
  __shared__ float mv[KS * 2];
  const int b = blockIdx.x, t = threadIdx.x;
  for (int i = t; i < KS * DD; i += 256) sn[i] = slots[b * KS * DD + i];
  __syncthreads();
  if (t < KS) {
    float m = 0.0f;
    for (int c = 0; c < DD; ++c) m += sn[t * DD + c];
    m *= (1.0f / DD);
    float va = 0.0f;
    for (int c = 0; c < DD; ++c) { float d = sn[t * DD + c] - m; va += d * d; }
    va *= (1.0f / DD);
    mv[t * 2] = m;
    mv[t * 2 + 1] = rsqrtf(va + LN_EPS);
  }
  __syncthreads();
  for (int i = t; i < KS * DD; i += 256) {
    int s = i >> 6, c = i & 63;
    sn[i] = (sn[i] - mv[s * 2]) * mv[s * 2 + 1] * lng[c] + lnb[c];
  }
  __syncthreads();
  for (int i = t; i < 16 * DD; i += 256) {
    int s = i >> 6, c = i & 63;
    float acc = 0.0f;
    if (s < KS)
      for (int d = 0; d < DD; ++d) acc += sn[s * DD + d] * Wq[c * DD + d];
    qpad[b * 16 * DD + i] = acc;   // rows 8..15 zero-padded
  }
}

// =====================================================================
// Kernel 3: streaming attention pass (WMMA f32 + async k staging).
//   scores C-layout puts all 8 slots of key-column n in ONE lane ->
//   softmax over slots is per-lane. Accumulate attn@v and row-sums,
//   combine across blocks with global f32 atomics.
// Grid: (8 chunks of 512 keys, B batches), 256 threads (8 waves).
// =====================================================================
#define CHUNK_N 512

__global__ __launch_bounds__(256) void attn_kernel(
    const float* __restrict__ qpad, const float* __restrict__ kbuf,
    const float* __restrict__ vbuf, float* __restrict__ upd_acc,
    float* __restrict__ rowsum) {
  __shared__ float ks[8 * 16 * XS_STR];   // per-wave k tile staging (34816 B)
  __shared__ float as[8 * 16 * 20];       // per-wave softmaxed tile (10240 B)
  const int t = threadIdx.x, lane = t & 31, wave = t >> 5;
  const int m16 = lane & 15, hi = lane >> 4, h2 = hi << 1;
  const int b = blockIdx.y;
  const long kvBase = (long)b * NN * DD;
  float* ksw = &ks[wave * 16 * XS_STR];
  float* asw = &as[wave * 16 * 20];
  const uint64_t kbase = (uint64_t)(kbuf + kvBase);

  // q A-fragments (slots padded to 16 rows)
  v2f qa[16];
  for (int s = 0; s < 16; ++s)
    qa[s] = *(const v2f*)&qpad[b * 16 * DD + m16 * DD + 4 * s + h2];

  v8f cu[4];
  for (int i = 0; i < 4; ++i) cu[i] = (v8f){};
  float rs[8];
  for (int j = 0; j < 8; ++j) rs[j] = 0.0f;

  const int nStart = blockIdx.x * CHUNK_N;
  for (int tt = 0; tt < CHUNK_N / (16 * 8); ++tt) {
    const int nb = nStart + (tt * 8 + wave) * 16;
    // async-stage 16x64 k tile (256 float4, 8 issues/lane, no VGPR data)
    for (int i = 0; i < 8; ++i) {
      int idx = i * 32 + lane, r = idx >> 4, c4 = idx & 15;
      async_load_b128(lds_addr32(&ksw[r * XS_STR + c4 * 4]),
                      (uint32_t)(((nb + r) * DD + c4 * 4) * 4), kbase);
    }
    async_wait0();   // tile is wave-private; ASYNCcnt is per-wave
    __syncthreads();
    // scores tile: C = q @ k^T  (16 slots x 16 keys)
    v8f c = {};
    for (int s = 0; s < 16; ++s) {
      v2f bk = *(const v2f*)&ksw[m16 * XS_STR + 4 * s + h2];
      c = wmma_f32(qa[s], bk, c);
    }
    // per-lane softmax over the 8 real slots (lanes 0-15 hold slots 0-7)
    float e[8], mx = -1e30f, sum = 0.0f;
    for (int j = 0; j < 8; ++j) { c[j] *= INV_SCALE; mx = fmaxf(mx, c[j]); }
    for (int j = 0; j < 8; ++j) { e[j] = expf(c[j] - mx); sum += e[j]; }
    float inv = 1.0f / sum;
    for (int j = 0; j < 8; ++j) {
      float av = e[j] * inv;
      // restage as A-matrix [slot][n_local]; pad rows 8..15 with zeros
      asw[(j + 8 * hi) * 20 + m16] = hi ? 0.0f : av;
      if (!hi) rs[j] += av;
    }
    __syncthreads();
    // updates += attn @ v   (Kdim = 16 keys, N = 64 dims in 4 tiles)
    for (int dt = 0; dt < 4; ++dt) {
      for (int s = 0; s < 4; ++s) {
        v2f af = *(const v2f*)&asw[m16 * 20 + 4 * s + h2];
        const int r0 = nb + 4 * s + h2, col = dt * 16 + m16;
        v2f bf;
        bf.x = vbuf[kvBase + (long)r0 * DD + col];
        bf.y = vbuf[kvBase + (long)(r0 + 1) * DD + col];
        cu[dt] = wmma_f32(af, bf, cu[dt]);
      }
    }
    __syncthreads();
  }

  // reduce row-sums across lanes 0-15 (lanes 16-31 contribute zeros)
  for (int j = 0; j < 8; ++j) {
    float v = rs[j];
    for (int off = 8; off >= 1; off >>= 1) v += __shfl_down(v, off, 32);
    if (lane == 0) atomicAdd(&rowsum[b * KS + j], v);
  }
  // updates partials: lanes 0-15 hold slots 0-7 (VGPR j = slot j)
  if (hi == 0) {
    for (int dt = 0; dt < 4; ++dt)
      for (int j = 0; j < 8; ++j)
        atomicAdd(&upd_acc[(b * KS + j) * DD + dt * 16 + m16], cu[dt][j]);
  }
}

// =====================================================================
// Kernel 4: finalize updates, GRUCell, LN + residual MLP -> new slots
// One block per (batch, slot), 64 threads (thread = output dim)
// =====================================================================
__global__ __launch_bounds__(64) void gru_mlp_kernel(
    const float* __restrict__ upd_acc, const float* __restrict__ rowsum,
    const float* __restrict__ slots_in, const float* __restrict__ wih,
    const float* __restrict__ whh, const float* __restrict__ bih,
    const float* __restrict__ bhh, const float* __restrict__ lng,
    const float* __restrict__ lnb, const float* __restrict__ w1,
    const float* __restrict__ b1, const float* __restrict__ w2,
    const float* __restrict__ b2, float* __restrict__ slots_out) {
  __shared__ float u[DD], p[DD], hn[DD], t1[4 * DD], red[DD];
  const int bk = blockIdx.x, t = threadIdx.x;
  const float inv = 1.0f / (rowsum[bk] + EPS_ATTN);
  u[t] = upd_acc[bk * DD + t] * inv;
  p[t] = slots_in[bk * DD + t];
  __syncthreads();

  float gr = bih[t], gz = bih[DD + t], gn = bih[2 * DD + t];
  float hr = bhh[t], hz = bhh[DD + t], hv = bhh[2 * DD + t];
  for (int c = 0; c < DD; ++c) {
    float uc = u[c], pc = p[c];
    gr += uc * wih[t * DD + c];
    gz += uc * wih[(DD + t) * DD + c];
    gn += uc * wih[(2 * DD + t) * DD + c];
    hr += pc * whh[t * DD + c];
    hz += pc * whh[(DD + t) * DD + c];
    hv += pc * whh[(2 * DD + t) * DD + c];
  }
  const float r = 1.0f / (1.0f + expf(-(gr + hr)));
  const float z = 1.0f / (1.0f + expf(-(gz + hz)));
  const float nn = tanhf(gn + r * hv);
  const float snew = (1.0f - z) * nn + z * p[t];

  // LayerNorm over the 64 dims (tree reductions)
  red[t] = snew;
  __syncthreads();
  for (int off = 32; off >= 1; off >>= 1) {
    if (t < off) red[t] += red[t + off];
    __syncthreads();
  }
  const float mean = red[0] * (1.0f / DD);
  __syncthreads();
  const float dv = snew - mean;
  red[t] = dv * dv;
  __syncthreads();
  for (int off = 32; off >= 1; off >>= 1) {
    if (t < off) red[t] += red[t + off];
    __syncthreads();
  }
  const float var = red[0] * (1.0f / DD);
  __syncthreads();
  hn[t] = dv * rsqrtf(var + LN_EPS) * lng[t] + lnb[t];
  __syncthreads();

  for (int o = t; o < 4 * DD; o += DD) {
    float acc = b1[o];
    for (int c = 0; c < DD; ++c) acc += hn[c] * w1[o * DD + c];
    t1[o] = fmaxf(acc, 0.0f);
  }
  __syncthreads();
  float acc = snew + b2[t];
  for (int o = 0; o < 4 * DD; ++o) acc += t1[o] * w2[t * 4 * DD + o];
  slots_out[bk * DD + t] = acc;
}

// =====================================================================
extern "C" void kernel_launch(void* const* d_in, const int* in_sizes, int n_in,
                              void* d_out, int out_size, void* d_ws,
                              size_t ws_size, hipStream_t stream) {
  const float* inputs  = (const float*)d_in[0];
  const float* noise   = (const float*)d_in[1];
  const float* slot_mu = (const float*)d_in[2];
  const float* slot_ls = (const float*)d_in[3];
  const float* Wq      = (const float*)d_in[4];
  const float* Wk      = (const float*)d_in[5];
  const float* Wv      = (const float*)d_in[6];
  const float* gru_wih = (const float*)d_in[7];
  const float* gru_whh = (const float*)d_in[8];
  const float* gru_bih = (const float*)d_in[9];
  const float* gru_bhh = (const float*)d_in[10];
  const float* ln_in_g = (const float*)d_in[11];
  const float* ln_in_b = (const float*)d_in[12];
  const float* ln_s_g  = (const float*)d_in[13];
  const float* ln_s_b  = (const float*)d_in[14];
  const float* mlp_lng = (const float*)d_in[15];
  const float* mlp_lnb = (const float*)d_in[16];
  const float* mlp_w1  = (const float*)d_in[17];
  const float* mlp_b1  = (const float*)d_in[18];
  const float* mlp_w2  = (const float*)d_in[19];
  const float* mlp_b2  = (const float*)d_in[20];

  // workspace carve-up (~269 MB)
  char* ws = (char*)d_ws;
  size_t off = 0;
  auto carve = [&](size_t bytes) {
    void* p = ws + off;
    off = (off + bytes + 255) & ~(size_t)255;
    return p;
  };
  const size_t SZ_KV = (size_t)BBATCH * NN * DD * sizeof(float);
  float* kbuf  = (float*)carve(SZ_KV);
  float* vbuf  = (float*)carve(SZ_KV);
  float* qpad  = (float*)carve((size_t)BBATCH * 16 * DD * sizeof(float));
  float* updA  = (float*)carve((size_t)BBATCH * KS * DD * sizeof(float));
  float* rsum  = (float*)carve((size_t)BBATCH * KS * sizeof(float));
  float* s0    = (float*)carve((size_t)BBATCH * KS * DD * sizeof(float));
  float* s1    = (float*)carve((size_t)BBATCH * KS * DD * sizeof(float));
  (void)ws_size; (void)in_sizes; (void)n_in; (void)out_size;

  const int nSlots = BBATCH * KS * DD;  // 65536
  init_slots_kernel<<<(nSlots + 255) / 256, 256, 0, stream>>>(
      noise, slot_mu, slot_ls, s0, nSlots);

  ln_kv_kernel<<<(BBATCH * NN) / LNKV_ROWS, 256, 0, stream>>>(
      inputs, ln_in_g, ln_in_b, Wk, Wv, kbuf, vbuf);

  float* cur = s0;
  float* nxt = s1;
  for (int it = 0; it < 3; ++it) {
    zero_kernel<<<(nSlots + 255) / 256, 256, 0, stream>>>(updA, nSlots);
    zero_kernel<<<(BBATCH * KS + 255) / 256, 256, 0, stream>>>(rsum, BBATCH * KS);
    q_kernel<<<BBATCH, 256, 0, stream>>>(cur, ln_s_g, ln_s_b, Wq, qpad);
    attn_kernel<<<dim3(NN / CHUNK_N, BBATCH), 256, 0, stream>>>(
        qpad, kbuf, vbuf, updA, rsum);
    float* outp = (it == 2) ? (float*)d_out : nxt;
    gru_mlp_kernel<<<BBATCH * KS, 64, 0, stream>>>(
        updA, rsum, cur, gru_wih, gru_whh, gru_bih, gru_bhh, mlp_lng, mlp_lnb,
        mlp_w1, mlp_b1, mlp_w2, mlp_b2, outp);
    float* tmp = cur;
    cur = nxt;
    nxt = tmp;
  }
}